// LinkPredictor_15470472200318
// MI455X (gfx1250) — compile-verified
//
#include <hip/hip_runtime.h>

typedef __attribute__((ext_vector_type(2))) float v2f;
typedef __attribute__((ext_vector_type(8))) float v8f;

#define DIM 128
#define SCAN_B 256

// ---------------- zero-fill ----------------

__global__ __launch_bounds__(256) void k_zero_i32(int* __restrict__ p, int n) {
  int i = blockIdx.x * 256 + threadIdx.x;
  if (i < n) p[i] = 0;
}

// ---------------- CSR build (by dst, incl. self loops) ----------------

// histogram of dst
__global__ __launch_bounds__(256) void k_count(const int* __restrict__ dst,
                                               int* __restrict__ cnt, int E, int total) {
  int i = blockIdx.x * 256 + threadIdx.x;
  if (i >= total) return;
  int t = (i < E) ? dst[i] : (i - E);
  atomicAdd(&cnt[t], 1);
}

// per-block exclusive scan
__global__ __launch_bounds__(SCAN_B) void k_scan_block(const int* __restrict__ cnt,
                                                       int* __restrict__ row_ptr,
                                                       int* __restrict__ blockSums, int N) {
  __shared__ int tmp[SCAN_B];
  int i = blockIdx.x * SCAN_B + threadIdx.x;
  int v = (i < N) ? cnt[i] : 0;
  tmp[threadIdx.x] = v;
  __syncthreads();
  for (int off = 1; off < SCAN_B; off <<= 1) {
    int add = (threadIdx.x >= (unsigned)off) ? tmp[threadIdx.x - off] : 0;
    __syncthreads();
    tmp[threadIdx.x] += add;
    __syncthreads();
  }
  if (i < N) row_ptr[i] = tmp[threadIdx.x] - v;  // exclusive
  if (threadIdx.x == SCAN_B - 1) blockSums[blockIdx.x] = tmp[SCAN_B - 1];
}

// single-block exclusive scan of block sums (nb <= 1024; N <= 262144)
__global__ __launch_bounds__(1024) void k_scan_sums(int* __restrict__ blockSums, int nb) {
  __shared__ int tmp[1024];
  int t = threadIdx.x;
  int v = (t < nb) ? blockSums[t] : 0;
  tmp[t] = v;
  __syncthreads();
  for (int off = 1; off < 1024; off <<= 1) {
    int add = (t >= off) ? tmp[t - off] : 0;
    __syncthreads();
    tmp[t] += add;
    __syncthreads();
  }
  if (t < nb) blockSums[t] = tmp[t] - v;  // exclusive
}

__global__ __launch_bounds__(SCAN_B) void k_scan_add(int* __restrict__ row_ptr,
                                                     const int* __restrict__ blockSums,
                                                     int N, int total) {
  int i = blockIdx.x * SCAN_B + threadIdx.x;
  if (i < N) row_ptr[i] += blockSums[blockIdx.x];
  if (i == 0) row_ptr[N] = total;
}

// drop edges into CSR slots (stores raw edge weight; normalized later)
__global__ __launch_bounds__(256) void k_fill_csr(const int* __restrict__ src,
                                                  const int* __restrict__ dst,
                                                  const float* __restrict__ ew,
                                                  const int* __restrict__ row_ptr,
                                                  int* __restrict__ fill,
                                                  int* __restrict__ csr_src,
                                                  float* __restrict__ csr_w,
                                                  int E, int total) {
  int i = blockIdx.x * 256 + threadIdx.x;
  if (i >= total) return;
  int s, t; float w;
  if (i < E) { s = src[i]; t = dst[i]; w = ew[i]; } else { s = t = i - E; w = 1.0f; }
  int pos = row_ptr[t] + atomicAdd(&fill[t], 1);
  csr_src[pos] = s;
  csr_w[pos] = w;
}

// deg[n] = sum of row weights (deterministic, no float atomics); dinv = rsqrt
__global__ __launch_bounds__(256) void k_deg_dinv(const float* __restrict__ csr_w,
                                                  const int* __restrict__ row_ptr,
                                                  float* __restrict__ dinv, int N) {
  int n = blockIdx.x * 256 + threadIdx.x;
  if (n >= N) return;
  float s = 0.0f;
  int end = row_ptr[n + 1];
  for (int j = row_ptr[n]; j < end; j++) s += csr_w[j];
  dinv[n] = s > 0.0f ? rsqrtf(s) : 0.0f;
}

// csr_w[j] = dinv[src]*w*dinv[dst]
__global__ __launch_bounds__(256) void k_csr_norm(const int* __restrict__ csr_src,
                                                  float* __restrict__ csr_w,
                                                  const int* __restrict__ row_ptr,
                                                  const float* __restrict__ dinv, int N) {
  int n = blockIdx.x * 256 + threadIdx.x;
  if (n >= N) return;
  float dn = dinv[n];
  int end = row_ptr[n + 1];
  for (int j = row_ptr[n]; j < end; j++)
    csr_w[j] = dinv[csr_src[j]] * csr_w[j] * dn;
}

// ---------------- WMMA f32 GEMM: H = X @ W, W staged in LDS ----------------
// Block = 8 waves = the 8 column tiles of one 16-row stripe; tM = blockIdx.x.
__global__ __launch_bounds__(256) void k_gemm_wmma(const float* __restrict__ X,
                                                   const float* __restrict__ W,
                                                   float* __restrict__ H) {
  __shared__ float lds_w[DIM * DIM];  // 64 KB: whole weight matrix
  for (int i = threadIdx.x; i < (DIM * DIM) / 4; i += 256)
    ((float4*)lds_w)[i] = ((const float4*)W)[i];
  __syncthreads();

  int lane = threadIdx.x & 31;
  int tN   = threadIdx.x >> 5;  // wave -> column tile
  int tM   = blockIdx.x;
  int row  = lane & 15;
  int kh   = (lane >> 4) << 1;  // lanes 0-15: K+{0,1}; lanes 16-31: K+{2,3}
  const float* xrow = X + (size_t)(tM * 16 + row) * DIM;
  const float* wcol = lds_w + tN * 16 + row;

  v8f acc = {};
#pragma unroll
  for (int kk = 0; kk < DIM; kk += 4) {
    v2f a = *(const v2f*)(xrow + kk + kh);  // A 16x4 fragment
    v2f b;
    b.x = wcol[(kk + kh) * DIM];            // B 4x16 fragment from LDS
    b.y = wcol[(kk + kh + 1) * DIM];
    acc = __builtin_amdgcn_wmma_f32_16x16x4_f32(false, a, false, b, (short)0, acc,
                                                false, false);
  }

  int orow = tM * 16 + ((lane >> 4) << 3);
  int ocol = tN * 16 + row;
  float* o = H + (size_t)orow * DIM + ocol;
#pragma unroll
  for (int r = 0; r < 8; r++) o[(size_t)r * DIM] = acc[r];
}

// scalar fallback for N % 16 rows (unused for N=100000)
__global__ __launch_bounds__(128) void k_gemm_tail(const float* __restrict__ X,
                                                   const float* __restrict__ W,
                                                   float* __restrict__ H,
                                                   int rowStart, int nRows) {
  if ((int)blockIdx.x >= nRows) return;
  int r = rowStart + blockIdx.x;
  int c = threadIdx.x;
  float s = 0.0f;
  for (int k = 0; k < DIM; k++) s += X[(size_t)r * DIM + k] * W[(size_t)k * DIM + c];
  H[(size_t)r * DIM + c] = s;
}

// ---------------- gather aggregation, fused bias + ReLU ----------------
// one wave per node; lane owns 4 features; no atomics
__global__ __launch_bounds__(256) void k_aggregate(const float* __restrict__ H,
                                                   const int* __restrict__ csr_src,
                                                   const float* __restrict__ csr_w,
                                                   const int* __restrict__ row_ptr,
                                                   const float* __restrict__ bias,
                                                   float* __restrict__ out, int N) {
  int lane = threadIdx.x & 31;
  int n = blockIdx.x * 8 + (threadIdx.x >> 5);
  if (n >= N) return;  // wave-uniform
  int d = lane * 4;
  float4 acc = *(const float4*)(bias + d);
  int end = row_ptr[n + 1];
  for (int j = row_ptr[n]; j < end; j++) {
    int s = csr_src[j];
    float w = csr_w[j];
    const float4 h = *(const float4*)(H + (size_t)s * DIM + d);
    acc.x += h.x * w; acc.y += h.y * w; acc.z += h.z * w; acc.w += h.w * w;
  }
  float4 r;
  r.x = fmaxf(acc.x, 0.0f); r.y = fmaxf(acc.y, 0.0f);
  r.z = fmaxf(acc.z, 0.0f); r.w = fmaxf(acc.w, 0.0f);
  *(float4*)(out + (size_t)n * DIM + d) = r;
}

// ---------------- folded final linear layer ----------------

__global__ void k_rowsum(const float* __restrict__ linW, const float* __restrict__ linb,
                         float* __restrict__ v, float* __restrict__ c) {
  int d = threadIdx.x;
  float s = 0.0f;
  for (int j = 0; j < DIM; j++) s += linW[(size_t)d * DIM + j];
  v[d] = s;
  if (d == 0) {
    float cc = 0.0f;
    for (int j = 0; j < DIM; j++) cc += linb[j];
    *c = cc;
  }
}

// res[e] = sum_d Z[a][d]*Z[b][d]*v[d] + c ; one wave per label edge
__global__ __launch_bounds__(256) void k_link(const float* __restrict__ Z,
                                              const int* __restrict__ la,
                                              const int* __restrict__ lb,
                                              const float* __restrict__ v,
                                              const float* __restrict__ c,
                                              float* __restrict__ res, int L) {
  int lane = threadIdx.x & 31;
  int e = blockIdx.x * 8 + (threadIdx.x >> 5);
  if (e >= L) return;  // wave-uniform
  int a = la[e], b = lb[e];
  const float4 xa = *(const float4*)(Z + (size_t)a * DIM + lane * 4);
  const float4 xb = *(const float4*)(Z + (size_t)b * DIM + lane * 4);
  const float4 vv = *(const float4*)(v + lane * 4);
  float p = xa.x * xb.x * vv.x + xa.y * xb.y * vv.y +
            xa.z * xb.z * vv.z + xa.w * xb.w * vv.w;
  for (int off = 16; off > 0; off >>= 1) p += __shfl_down(p, off, 32);
  if (lane == 0) res[e] = p + *c;
}

// ---------------- launcher ----------------

extern "C" void kernel_launch(void* const* d_in, const int* in_sizes, int n_in,
                              void* d_out, int out_size, void* d_ws, size_t ws_size,
                              hipStream_t stream) {
  const float* x    = (const float*)d_in[0];
  const int*   ei   = (const int*)d_in[1];
  const float* ew   = (const float*)d_in[2];
  const int*   eli  = (const int*)d_in[3];
  const float* W1   = (const float*)d_in[4];
  const float* b1   = (const float*)d_in[5];
  const float* W2   = (const float*)d_in[6];
  const float* b2   = (const float*)d_in[7];
  const float* linW = (const float*)d_in[8];
  const float* linb = (const float*)d_in[9];
  float* out = (float*)d_out;

  const int N = in_sizes[0] / DIM;
  const int E = in_sizes[1] / 2;
  const int L = in_sizes[3] / 2;
  const int T = E + N;  // edges incl. self loops
  const int* src = ei;
  const int* dstp = ei + E;
  const int* la = eli;
  const int* lb = eli + L;
  const int nb = (N + SCAN_B - 1) / SCAN_B;  // scan blocks (<=1024 for N<=262144)

  // workspace carve-up (64-element aligned regions, 4-byte units)
  auto align64 = [](size_t v) { return (v + 63) & ~(size_t)63; };
  char* wsb = (char*)d_ws;
  size_t off = 0;
  int*   cnt      = (int*)(wsb + off * 4); off = align64(off + (size_t)N);
  int*   fill     = (int*)(wsb + off * 4); off = align64(off + (size_t)N);
  int*   row_ptr  = (int*)(wsb + off * 4); off = align64(off + (size_t)N + 1);
  int*   bsums    = (int*)(wsb + off * 4); off = align64(off + (size_t)nb);
  int*   csr_src  = (int*)(wsb + off * 4); off = align64(off + (size_t)T);
  float* csr_w    = (float*)(wsb + off * 4); off = align64(off + (size_t)T);
  float* dinv     = (float*)(wsb + off * 4); off = align64(off + (size_t)N);
  float* bufA     = (float*)(wsb + off * 4); off = align64(off + (size_t)N * DIM);
  float* bufB     = (float*)(wsb + off * 4); off = align64(off + (size_t)N * DIM);
  float* vvec     = (float*)(wsb + off * 4); off = align64(off + DIM);
  float* cval     = (float*)(wsb + off * 4);

  const int tilesM = N / 16;
  const int remRows = N - tilesM * 16;
  dim3 blk(256);
  int gN = (N + 255) / 256;
  int gT = (T + 255) / 256;

  // ---- build CSR by dst (reused by both layers) ----
  k_zero_i32<<<gN, blk, 0, stream>>>(cnt, N);
  k_count<<<gT, blk, 0, stream>>>(dstp, cnt, E, T);
  k_scan_block<<<nb, SCAN_B, 0, stream>>>(cnt, row_ptr, bsums, N);
  k_scan_sums<<<1, 1024, 0, stream>>>(bsums, nb);
  k_scan_add<<<nb, SCAN_B, 0, stream>>>(row_ptr, bsums, N, T);
  k_zero_i32<<<gN, blk, 0, stream>>>(fill, N);
  k_fill_csr<<<gT, blk, 0, stream>>>(src, dstp, ew, row_ptr, fill, csr_src, csr_w, E, T);
  k_deg_dinv<<<gN, blk, 0, stream>>>(csr_w, row_ptr, dinv, N);
  k_csr_norm<<<gN, blk, 0, stream>>>(csr_src, csr_w, row_ptr, dinv, N);

  // ---- layer 1: bufA = x@W1 ; bufB = relu(gather(bufA) + b1) ----
  if (tilesM > 0) k_gemm_wmma<<<tilesM, blk, 0, stream>>>(x, W1, bufA);
  if (remRows > 0) k_gemm_tail<<<remRows, DIM, 0, stream>>>(x, W1, bufA, tilesM * 16, remRows);
  k_aggregate<<<(N + 7) / 8, blk, 0, stream>>>(bufA, csr_src, csr_w, row_ptr, b1, bufB, N);

  // ---- layer 2: bufA = bufB@W2 ; bufB = relu(gather(bufA) + b2) ----
  if (tilesM > 0) k_gemm_wmma<<<tilesM, blk, 0, stream>>>(bufB, W2, bufA);
  if (remRows > 0) k_gemm_tail<<<remRows, DIM, 0, stream>>>(bufB, W2, bufA, tilesM * 16, remRows);
  k_aggregate<<<(N + 7) / 8, blk, 0, stream>>>(bufA, csr_src, csr_w, row_ptr, b2, bufB, N);

  // ---- link prediction: res = (Z[la]*Z[lb]) . rowsum(lin_W) + sum(lin_b) ----
  k_rowsum<<<1, DIM, 0, stream>>>(linW, linb, vvec, cval);
  k_link<<<(L + 7) / 8, blk, 0, stream>>>(bufB, la, lb, vvec, cval, out, L);
}